// Memory_23905787969627
// MI455X (gfx1250) — compile-verified
//
#include <hip/hip_runtime.h>
#include <math.h>

#define B_   256
#define M_   2048
#define C_   64
#define RH_  4
#define IN_  512
#define IF_  461
#define EPSF   1e-6f
#define DELTAF 1e-6f

typedef __attribute__((ext_vector_type(2))) float v2f;
typedef __attribute__((ext_vector_type(8))) float v8f;

// ---------------- workspace layout (floats) ----------------
static constexpr size_t O_ITF = 0;                               // B*IF
static constexpr size_t O_RK  = O_ITF + (size_t)B_*IF_;          // B*4*64 read keys (tanh)
static constexpr size_t O_RS  = O_RK  + (size_t)B_*RH_*C_;       // B*4 read strengths
static constexpr size_t O_WK  = O_RS  + (size_t)B_*RH_;          // B*64 write key (tanh)
static constexpr size_t O_WSV = O_WK  + (size_t)B_*C_;           // B write strength
static constexpr size_t O_EV  = O_WSV + (size_t)B_;              // B*64 erase
static constexpr size_t O_WV  = O_EV  + (size_t)B_*C_;           // B*64 write vec
static constexpr size_t O_FG  = O_WV  + (size_t)B_*C_;           // B*4 free gates
static constexpr size_t O_AG  = O_FG  + (size_t)B_*RH_;          // B*3 alloc gate (softmaxed)
static constexpr size_t O_WG  = O_AG  + (size_t)B_*3;            // B write gate
static constexpr size_t O_WSC = O_WG  + (size_t)B_;              // B write key scale
static constexpr size_t O_RSC = O_WSC + (size_t)B_;              // B*4 read key scales
static constexpr size_t O_U2  = O_RSC + (size_t)B_*RH_;          // B*M
static constexpr size_t O_WCW = O_U2  + (size_t)B_*M_;           // B*M (logits -> softmax in place)
static constexpr size_t O_AL  = O_WCW + (size_t)B_*M_;           // B*M allocation
static constexpr size_t O_WW  = O_AL  + (size_t)B_*M_;           // B*M write weighting
static constexpr size_t O_RL  = O_WW  + (size_t)B_*M_;           // B*4*M read logits -> rw in place

// ---------------- helpers ----------------
__device__ __forceinline__ float softplus_f(float x) {
  return (x > 0.f) ? (x + log1pf(expf(-x))) : log1pf(expf(x));
}
__device__ __forceinline__ float sigmoid_f(float x) { return 1.f / (1.f + expf(-x)); }
__device__ __forceinline__ float wred32(float v) {
#pragma unroll
  for (int off = 16; off > 0; off >>= 1) v += __shfl_xor(v, off, 32);
  return v;
}

// ---------------- K1: itf = xi @ W^T via V_WMMA_F32_16X16X4_F32 ----------------
// 8 waves per block; each wave computes one 16x16 tile of (b x IFACE).
// A (16x4 f32): lane<16 holds M=lane K={k,k+1}; lane>=16 holds M=lane-16 K={k+2,k+3}.
// B (4x16 f32): symmetric — VGPR0/1 hold rows K={0,1} (lanes 0-15) / K={2,3} (lanes 16-31).
// Out-of-range N columns: pointer clamped (loads stay in-bounds); their lanes produce
// garbage accumulators but stores are guarded, and columns are independent in D=A*B+C.
__global__ __launch_bounds__(256) void k_itf_gemm(const float* __restrict__ xi,
                                                  const float* __restrict__ W,
                                                  float* __restrict__ itf) {
  const int tid  = threadIdx.x;
  const int wave = tid >> 5;
  const int lane = tid & 31;
  const int half = lane >> 4;
  const int l15  = lane & 15;
  const int m0 = blockIdx.x * 16;
  const int n0 = (blockIdx.y * 8 + wave) * 16;
  const int ncol = n0 + l15;
  const int ncl  = (ncol < IF_) ? ncol : (IF_ - 1);
  const float* arow = xi + (size_t)(m0 + l15) * IN_ + half * 2;
  const float* brow = W  + (size_t)ncl * IN_ + half * 2;
  v8f acc = {};
  for (int k = 0; k < IN_; k += 4) {
    v2f a = *(const v2f*)(arow + k);
    v2f b = *(const v2f*)(brow + k);
    acc = __builtin_amdgcn_wmma_f32_16x16x4_f32(false, a, false, b,
                                                (short)0, acc, false, false);
  }
  if (ncol < IF_) {
#pragma unroll
    for (int i = 0; i < 8; ++i)
      itf[(size_t)(m0 + half * 8 + i) * IF_ + ncol] = acc[i];
  }
}

// ---------------- K2: GroupNorm(1,C) + activations ----------------
__global__ __launch_bounds__(512) void k_gnorm(const float* __restrict__ itf,
    const float* __restrict__ gamma, const float* __restrict__ beta,
    float* __restrict__ rk, float* __restrict__ rs, float* __restrict__ wk,
    float* __restrict__ wsv, float* __restrict__ ev, float* __restrict__ wv,
    float* __restrict__ fg, float* __restrict__ ag, float* __restrict__ wg) {
  __shared__ float red[512];
  __shared__ float ag3[3];
  const int b = blockIdx.x;
  const int j = threadIdx.x;
  float val = (j < IF_) ? itf[(size_t)b * IF_ + j] : 0.f;
  red[j] = val;
  __syncthreads();
  for (int s = 256; s > 0; s >>= 1) { if (j < s) red[j] += red[j + s]; __syncthreads(); }
  float mean = red[0] * (1.f / IF_);
  __syncthreads();
  red[j] = val * val;
  __syncthreads();
  for (int s = 256; s > 0; s >>= 1) { if (j < s) red[j] += red[j + s]; __syncthreads(); }
  float var  = red[0] * (1.f / IF_) - mean * mean;
  float rstd = rsqrtf(var + 1e-5f);
  if (j < IF_) {
    float y = (val - mean) * rstd * gamma[j] + beta[j];
    if (j < 256)       rk[(size_t)b * 256 + j] = tanhf(y);
    else if (j < 260)  rs[b * 4 + (j - 256)] = softplus_f(y);
    else if (j < 324)  wk[b * 64 + (j - 260)] = tanhf(y);
    else if (j == 324) wsv[b] = softplus_f(y);
    else if (j < 389)  ev[b * 64 + (j - 325)] = sigmoid_f(y);
    else if (j < 453)  wv[b * 64 + (j - 389)] = tanhf(y);
    else if (j < 457)  fg[b * 4 + (j - 453)] = sigmoid_f(y);
    else if (j < 460)  ag3[j - 457] = y;
    else               wg[b] = sigmoid_f(y);
  }
  __syncthreads();
  if (j == 0) {
    float a0 = ag3[0], a1 = ag3[1], a2 = ag3[2];
    float mx = fmaxf(a0, fmaxf(a1, a2));
    float e0 = expf(a0 - mx), e1 = expf(a1 - mx), e2 = expf(a2 - mx);
    float inv = 1.f / (e0 + e1 + e2);
    ag[b * 3 + 0] = e0 * inv; ag[b * 3 + 1] = e1 * inv; ag[b * 3 + 2] = e2 * inv;
  }
}

// ---------------- K3: key-norm scales (strength / (||key||+eps)) ----------------
__global__ __launch_bounds__(32) void k_scales(const float* __restrict__ wk,
    const float* __restrict__ rk, const float* __restrict__ wsv,
    const float* __restrict__ rs, float* __restrict__ wscale,
    float* __restrict__ rscale) {
  const int b = blockIdx.x;
  const int lane = threadIdx.x;
  const int c0 = 2 * lane;
  float a0 = wk[b * 64 + c0], a1 = wk[b * 64 + c0 + 1];
  float s = wred32(a0 * a0 + a1 * a1);
  if (lane == 0) wscale[b] = wsv[b] / (sqrtf(s) + EPSF);
#pragma unroll
  for (int r = 0; r < 4; ++r) {
    float k0 = rk[(size_t)b * 256 + r * 64 + c0];
    float k1 = rk[(size_t)b * 256 + r * 64 + c0 + 1];
    float sr = wred32(k0 * k0 + k1 * k1);
    if (lane == 0) rscale[b * 4 + r] = rs[b * 4 + r] / (sqrtf(sr) + EPSF);
  }
}

// ---------------- K4: usage update -> u2 ----------------
__global__ __launch_bounds__(256) void k_u2(const float* __restrict__ usage,
    const float* __restrict__ wwin, const float* __restrict__ rwin,
    const float* __restrict__ fg, float* __restrict__ u2) {
  int i = blockIdx.x * 256 + threadIdx.x;
  int b = i >> 11, mi = i & (M_ - 1);
  float us = usage[i];
  float u = us + (1.f - us) * wwin[i];
  float psi = 1.f;
#pragma unroll
  for (int r = 0; r < 4; ++r)
    psi *= (1.f - fg[b * 4 + r] * rwin[(size_t)(b * 4 + r) * M_ + mi]);
  u *= psi;
  u2[i] = DELTAF + (1.f - DELTAF) * u;
}

// ---------------- K5: write content logits (pass 1 over memory) ----------------
__global__ __launch_bounds__(256) void k_wlogits(const float* __restrict__ mem,
    const float* __restrict__ wk, const float* __restrict__ wscale,
    float* __restrict__ wlog) {
  const int b = blockIdx.x;
  const int tid = threadIdx.x;
  const int wave = tid >> 5, lane = tid & 31;
  const int c0 = 2 * lane;
  const float k0 = wk[b * 64 + c0], k1 = wk[b * 64 + c0 + 1];
  const float wsc = wscale[b];
  const int base = blockIdx.y * 64 + wave * 8;
  for (int it = 0; it < 8; ++it) {
    int row = base + it;
    const float* mr = mem + ((size_t)b * M_ + row) * C_ + c0;
    v2f x = *(const v2f*)mr;
    float d = wred32(x[0] * k0 + x[1] * k1);
    float n = wred32(x[0] * x[0] + x[1] * x[1]);
    if (lane == 0) wlog[(size_t)b * M_ + row] = d / (sqrtf(n) + EPSF) * wsc;
  }
}

// ---------------- row softmax over M (in place) ----------------
__global__ __launch_bounds__(256) void k_softmax_rows(float* __restrict__ buf) {
  __shared__ float lds[M_];
  __shared__ float red[256];
  float* p = buf + (size_t)blockIdx.x * M_;
  const int tid = threadIdx.x;
  float lmx = -3.0e38f;
  for (int i = tid; i < M_; i += 256) { float v = p[i]; lds[i] = v; lmx = fmaxf(lmx, v); }
  red[tid] = lmx; __syncthreads();
  for (int s = 128; s > 0; s >>= 1) { if (tid < s) red[tid] = fmaxf(red[tid], red[tid + s]); __syncthreads(); }
  float mx = red[0]; __syncthreads();
  float lsum = 0.f;
  for (int i = tid; i < M_; i += 256) { float e = expf(lds[i] - mx); lds[i] = e; lsum += e; }
  red[tid] = lsum; __syncthreads();
  for (int s = 128; s > 0; s >>= 1) { if (tid < s) red[tid] += red[tid + s]; __syncthreads(); }
  float inv = 1.f / red[0];
  for (int i = tid; i < M_; i += 256) p[i] = lds[i] * inv;
}

// ---------------- K6: allocation = sort + cumprod + scatter, in LDS ----------------
__global__ __launch_bounds__(1024) void k_alloc(const float* __restrict__ u2,
                                                float* __restrict__ alloc) {
  __shared__ float sv[M_];
  __shared__ unsigned short si[M_];
  __shared__ float cp[M_];
  const int b = blockIdx.x;
  const int tid = threadIdx.x;
  for (int i = tid; i < M_; i += 1024) { sv[i] = u2[(size_t)b * M_ + i]; si[i] = (unsigned short)i; }
  // bitonic sort ascending with (value, index) key (stable-equivalent; ties broken by index)
  for (int k = 2; k <= M_; k <<= 1) {
    for (int j = k >> 1; j > 0; j >>= 1) {
      __syncthreads();
      for (int t = tid; t < M_; t += 1024) {
        int l = t ^ j;
        if (l > t) {
          bool up = ((t & k) == 0);
          float va = sv[t], vb = sv[l];
          unsigned short ia = si[t], ib = si[l];
          bool agtb = (va > vb) || (va == vb && ia > ib);
          if (up == agtb) { sv[t] = vb; sv[l] = va; si[t] = ib; si[l] = ia; }
        }
      }
    }
  }
  __syncthreads();
  for (int i = tid; i < M_; i += 1024) cp[i] = sv[i];
  __syncthreads();
  // inclusive product scan (Hillis-Steele)
  const int i0 = tid, i1 = tid + 1024;
  for (int off = 1; off < M_; off <<= 1) {
    float v0 = (i0 >= off) ? cp[i0 - off] : 1.f;
    float v1 = (i1 >= off) ? cp[i1 - off] : 1.f;
    __syncthreads();
    cp[i0] *= v0; cp[i1] *= v1;
    __syncthreads();
  }
  for (int i = tid; i < M_; i += 1024) {
    float excl = (i == 0) ? 1.f : cp[i - 1];
    alloc[(size_t)b * M_ + si[i]] = (1.f - sv[i]) * excl;
  }
}

// ---------------- K7: write weighting ww ----------------
__global__ __launch_bounds__(256) void k_ww(const float* __restrict__ rwin,
    const float* __restrict__ alloc, const float* __restrict__ wcw,
    const float* __restrict__ ag, const float* __restrict__ wg,
    float* __restrict__ ww) {
  int i = blockIdx.x * 256 + threadIdx.x;
  int b = i >> 11, mi = i & (M_ - 1);
  float mr = 0.f;
#pragma unroll
  for (int r = 0; r < 4; ++r) mr += rwin[(size_t)(b * 4 + r) * M_ + mi];
  mr *= 0.25f;
  float fw = ag[b * 3 + 0] * mr + ag[b * 3 + 1] * alloc[i] + ag[b * 3 + 2] * wcw[i];
  ww[i] = wg[b] * fw;
}

// ---------------- K8: read logits (pass 2, erase/add fused on the fly) ----------------
__global__ __launch_bounds__(256) void k_rlogits(const float* __restrict__ mem,
    const float* __restrict__ ww, const float* __restrict__ ev,
    const float* __restrict__ wv, const float* __restrict__ rk,
    const float* __restrict__ rscale, float* __restrict__ rlog) {
  const int b = blockIdx.x;
  const int tid = threadIdx.x;
  const int wave = tid >> 5, lane = tid & 31;
  const int c0 = 2 * lane;
  const float e0 = ev[b * 64 + c0], e1 = ev[b * 64 + c0 + 1];
  const float w0 = wv[b * 64 + c0], w1 = wv[b * 64 + c0 + 1];
  float ka[4], kb[4], rsc[4];
#pragma unroll
  for (int r = 0; r < 4; ++r) {
    ka[r] = rk[(size_t)b * 256 + r * 64 + c0];
    kb[r] = rk[(size_t)b * 256 + r * 64 + c0 + 1];
    rsc[r] = rscale[b * 4 + r];
  }
  const int base = blockIdx.y * 64 + wave * 8;
  for (int it = 0; it < 8; ++it) {
    int row = base + it;
    float wwv = ww[(size_t)b * M_ + row];
    const float* mr = mem + ((size_t)b * M_ + row) * C_ + c0;
    v2f xm = *(const v2f*)mr;
    float x0 = xm[0] * (1.f - wwv * e0) + wwv * w0;
    float x1 = xm[1] * (1.f - wwv * e1) + wwv * w1;
    float dn = wred32(x0 * x0 + x1 * x1);
    float inv = 1.f / (sqrtf(dn) + EPSF);
#pragma unroll
    for (int r = 0; r < 4; ++r) {
      float d = wred32(x0 * ka[r] + x1 * kb[r]);
      if (lane == 0) rlog[((size_t)b * 4 + r) * M_ + row] = d * inv * rsc[r];
    }
  }
}

// ---------------- K9: read vectors (pass 3, erase/add fused, 8-row LDS staging) ----------------
__global__ __launch_bounds__(256) void k_readvec(const float* __restrict__ mem,
    const float* __restrict__ ww, const float* __restrict__ ev,
    const float* __restrict__ wv, const float* __restrict__ rw,
    float* __restrict__ out) {
  __shared__ float rowbuf[512];
  const int b = blockIdx.x;
  const int tid = threadIdx.x;
  const int r = tid >> 6, cc = tid & 63;
  const float e = ev[b * 64 + cc];
  const float wvc = wv[b * 64 + cc];
  float acc = 0.f;
  for (int m0 = 0; m0 < M_; m0 += 8) {
#pragma unroll
    for (int p = 0; p < 2; ++p) {
      int row = m0 + p * 4 + (tid >> 6);
      float wwv = ww[(size_t)b * M_ + row];
      float x = mem[((size_t)b * M_ + row) * C_ + cc];
      rowbuf[p * 256 + tid] = x * (1.f - wwv * e) + wwv * wvc;
    }
    __syncthreads();
    const float* rwb = rw + ((size_t)b * 4 + r) * M_ + m0;
#pragma unroll
    for (int jj = 0; jj < 8; ++jj) acc += rwb[jj] * rowbuf[jj * 64 + cc];
    __syncthreads();
  }
  out[((size_t)b * 4 + r) * C_ + cc] = acc;
}

// ---------------- host launch ----------------
extern "C" void kernel_launch(void* const* d_in, const int* in_sizes, int n_in,
                              void* d_out, int out_size, void* d_ws, size_t ws_size,
                              hipStream_t stream) {
  const float* xi    = (const float*)d_in[0];
  const float* W     = (const float*)d_in[1];
  const float* gamma = (const float*)d_in[2];
  const float* beta  = (const float*)d_in[3];
  const float* mem   = (const float*)d_in[4];
  const float* rwin  = (const float*)d_in[5];
  const float* wwin  = (const float*)d_in[6];
  const float* usage = (const float*)d_in[7];
  float* out = (float*)d_out;
  float* ws = (float*)d_ws;

  float* itf    = ws + O_ITF;
  float* rk     = ws + O_RK;
  float* rs     = ws + O_RS;
  float* wk     = ws + O_WK;
  float* wsv    = ws + O_WSV;
  float* ev     = ws + O_EV;
  float* wv     = ws + O_WV;
  float* fg     = ws + O_FG;
  float* ag     = ws + O_AG;
  float* wg     = ws + O_WG;
  float* wscale = ws + O_WSC;
  float* rscale = ws + O_RSC;
  float* u2     = ws + O_U2;
  float* wcw    = ws + O_WCW;
  float* alloc  = ws + O_AL;
  float* wwv    = ws + O_WW;
  float* rlog   = ws + O_RL;

  k_itf_gemm<<<dim3(16, 4), 256, 0, stream>>>(xi, W, itf);
  k_gnorm<<<B_, 512, 0, stream>>>(itf, gamma, beta, rk, rs, wk, wsv, ev, wv, fg, ag, wg);
  k_scales<<<B_, 32, 0, stream>>>(wk, rk, wsv, rs, wscale, rscale);
  k_u2<<<(B_ * M_) / 256, 256, 0, stream>>>(usage, wwin, rwin, fg, u2);
  k_wlogits<<<dim3(B_, M_ / 64), 256, 0, stream>>>(mem, wk, wscale, wcw);
  k_softmax_rows<<<B_, 256, 0, stream>>>(wcw);
  k_alloc<<<B_, 1024, 0, stream>>>(u2, alloc);
  k_ww<<<(B_ * M_) / 256, 256, 0, stream>>>(rwin, alloc, wcw, ag, wg, wwv);
  k_rlogits<<<dim3(B_, M_ / 64), 256, 0, stream>>>(mem, wwv, ev, wv, rk, rscale, rlog);
  k_softmax_rows<<<B_ * RH_, 256, 0, stream>>>(rlog);
  k_readvec<<<B_, 256, 0, stream>>>(mem, wwv, ev, wv, rlog, out);
}